// TransformerBlockQuantum_65481071407927
// MI455X (gfx1250) — compile-verified
//
#include <hip/hip_runtime.h>
#include <math.h>

typedef __attribute__((ext_vector_type(2))) float v2f;
typedef __attribute__((ext_vector_type(8))) float v8f;

// Fixed problem dims (B,S,E,F) = (8,256,256,1024), N_ATTN = N_FFN = 8
constexpr int Esz   = 256;
constexpr int Fsz   = 1024;
constexpr int Mrows = 8 * 256;            // B*S = 2048 token rows
constexpr int QROWS = Mrows * (Esz / 8);  // 65536 quantum rows of 8

// ---------------------------------------------------------------------------
// Kernel A: exact closed form of the RX + CNOT-ring + MeasureZ layer.
// ez[0] = c1*...*c7 ; ez[w] = c0*...*cw (w>=1), c_j = cos(v_j + theta_j)
// ---------------------------------------------------------------------------
__global__ void ez_kernel(const float* __restrict__ src,
                          const float* __restrict__ theta,
                          float* __restrict__ ez) {
  int r = blockIdx.x * blockDim.x + threadIdx.x;
  if (r >= QROWS) return;
  const float* v = src + (size_t)r * 8;
  float c[8];
#pragma unroll
  for (int j = 0; j < 8; ++j) c[j] = cosf(v[j] + theta[j]);
  float o[8];
  float p = c[0];
#pragma unroll
  for (int w = 1; w < 8; ++w) { p *= c[w]; o[w] = p; }
  float s = c[1];
#pragma unroll
  for (int j = 2; j < 8; ++j) s *= c[j];
  o[0] = s;
  float* dst = ez + (size_t)r * 8;
#pragma unroll
  for (int j = 0; j < 8; ++j) dst[j] = o[j];
}

// ---------------------------------------------------------------------------
// Fused GEMM (out = A[2048,K] @ W[256,K]^T + bias) + residual + LayerNorm.
// One block = 16 waves = one 16-row stripe across all 256 output columns.
// Wave wv computes 16x16 tile (cols wv*16..) with V_WMMA_F32_16X16X4_F32.
// fp32 WMMA fragment mapping (ISA 7.12.2): lo=lane&15, hi=lane>>4.
//   A frag: lane -> A[lo][kb+2*hi], A[lo][kb+2*hi+1]
//   B frag: lane -> B[kb+2*hi][lo] = W[colN][kb+2*hi], W[colN][kb+2*hi+1]
//   D frag: vgpr v -> D[v+8*hi][lo]
// ---------------------------------------------------------------------------
template <int K>
__global__ __launch_bounds__(512)
void gemm_resid_ln(const float* __restrict__ A,
                   const float* __restrict__ W,
                   const float* __restrict__ bias,
                   const float* __restrict__ resid,
                   const float* __restrict__ gamma,
                   const float* __restrict__ beta,
                   float* __restrict__ out) {
  __shared__ float tile[16 * 256];

  const int lane = threadIdx.x & 31;
  const int wv   = threadIdx.x >> 5;        // 0..15 : output column tile
  const int lo   = lane & 15;
  const int hi   = lane >> 4;
  const int rowA = blockIdx.x * 16 + lo;    // A row carried by this lane
  const int colN = wv * 16 + lo;            // global output column

  const float* ap = A + (size_t)rowA * K + 2 * hi;
  const float* bp = W + (size_t)colN * K + 2 * hi;

  v8f acc = {};
  for (int kb = 0; kb < K; kb += 4) {
    v2f a = *(const v2f*)(ap + kb);
    v2f b = *(const v2f*)(bp + kb);
    acc = __builtin_amdgcn_wmma_f32_16x16x4_f32(
        /*neg_a=*/false, a, /*neg_b=*/false, b,
        /*c_mod=*/(short)0, acc, /*reuse_a=*/false, /*reuse_b=*/false);
  }

  // Scatter D tile to LDS with bias + residual fused in.
  const float bn = bias[colN];
#pragma unroll
  for (int vi = 0; vi < 8; ++vi) {
    int ml = vi + 8 * hi;
    int mg = blockIdx.x * 16 + ml;
    tile[ml * 256 + colN] = acc[vi] + bn + resid[(size_t)mg * 256 + colN];
  }
  __syncthreads();

  // LayerNorm: wave wv normalizes local row wv (256 values, 8 per lane).
  const int r  = wv;
  const int mg = blockIdx.x * 16 + r;
  float vals[8];
  float s = 0.f, s2 = 0.f;
#pragma unroll
  for (int j = 0; j < 8; ++j) {
    float x = tile[r * 256 + lane + 32 * j];
    vals[j] = x;
    s  += x;
    s2 += x * x;
  }
#pragma unroll
  for (int m = 16; m >= 1; m >>= 1) {
    s  += __shfl_xor(s,  m, 32);
    s2 += __shfl_xor(s2, m, 32);
  }
  const float mu  = s * (1.0f / 256.0f);
  const float var = s2 * (1.0f / 256.0f) - mu * mu;
  const float rs  = rsqrtf(var + 1e-5f);
#pragma unroll
  for (int j = 0; j < 8; ++j) {
    int c = lane + 32 * j;
    out[(size_t)mg * 256 + c] = (vals[j] - mu) * rs * gamma[c] + beta[c];
  }
}

// ---------------------------------------------------------------------------
// FFN first layer: qf = cos(x[:, :8]) * cos(theta_ffn); h = relu(qf @ w1^T + b1)
// K=8 -> VALU FMAs (WMMA would waste 75% of the K dimension).
// One block covers 256 f-columns of one token row.
// ---------------------------------------------------------------------------
__global__ void ffn1_kernel(const float* __restrict__ x,
                            const float* __restrict__ theta_ffn,
                            const float* __restrict__ w1,
                            const float* __restrict__ b1,
                            float* __restrict__ h) {
  __shared__ float qf[8];
  const int row = blockIdx.x >> 2;
  const int f   = ((blockIdx.x & 3) << 8) + threadIdx.x;
  if (threadIdx.x < 8) {
    int k = threadIdx.x;
    qf[k] = cosf(x[(size_t)row * Esz + k]) * cosf(theta_ffn[k]);
  }
  __syncthreads();
  const float* wr = w1 + (size_t)f * 8;
  float acc = b1[f];
#pragma unroll
  for (int k = 0; k < 8; ++k) acc = fmaf(qf[k], wr[k], acc);
  h[(size_t)row * Fsz + f] = fmaxf(acc, 0.0f);
}

// ---------------------------------------------------------------------------
extern "C" void kernel_launch(void* const* d_in, const int* in_sizes, int n_in,
                              void* d_out, int out_size, void* d_ws, size_t ws_size,
                              hipStream_t stream) {
  const float* src        = (const float*)d_in[0];
  const float* theta_attn = (const float*)d_in[1];
  const float* theta_ffn  = (const float*)d_in[2];
  const float* w_combine  = (const float*)d_in[3];
  const float* b_combine  = (const float*)d_in[4];
  const float* w1         = (const float*)d_in[5];
  const float* b1         = (const float*)d_in[6];
  const float* w2         = (const float*)d_in[7];
  const float* b2         = (const float*)d_in[8];
  const float* g1         = (const float*)d_in[9];
  const float* be1        = (const float*)d_in[10];
  const float* g2         = (const float*)d_in[11];
  const float* be2        = (const float*)d_in[12];
  float* out = (float*)d_out;

  char* ws = (char*)d_ws;
  float* ez   = (float*)(ws);                              // 2048*256 f32 (2 MB)
  float* xbuf = (float*)(ws + (size_t)Mrows * Esz * 4);    // 2048*256 f32 (2 MB)
  float* hbuf = (float*)(ws + (size_t)2 * Mrows * Esz * 4);// 2048*1024 f32 (8 MB)

  // 1) quantum attention expectations
  ez_kernel<<<QROWS / 256, 256, 0, stream>>>(src, theta_attn, ez);
  // 2) attn combine GEMM (K=256) + residual + LN1 -> x
  gemm_resid_ln<Esz><<<Mrows / 16, 512, 0, stream>>>(
      ez, w_combine, b_combine, src, g1, be1, xbuf);
  // 3) quantum-encoded FFN up-projection (K=8) + ReLU -> h
  ffn1_kernel<<<Mrows * 4, 256, 0, stream>>>(xbuf, theta_ffn, w1, b1, hbuf);
  // 4) FFN down GEMM (K=1024) + residual + LN2 -> out
  gemm_resid_ln<Fsz><<<Mrows / 16, 512, 0, stream>>>(
      hbuf, w2, b2, xbuf, g2, be2, out);
}